// MCPInitEmbedding_37752762532212
// MI455X (gfx1250) — compile-verified
//
#include <hip/hip_runtime.h>

// MI455X / gfx1250, wave32.
// out[b,s,d] = S[b,s]*w[d] + SET_SZ*bias[d],  S[b,s] = sum_k weights[b, m[b,s,k]]
// Linearity of the embedding collapses the [B,N_SETS,SET_SZ,D] gather into a
// [B,N_SETS] gather-reduce plus a rank-1 expansion done on the matrix pipe.

typedef __attribute__((ext_vector_type(2))) float v2f;
typedef __attribute__((ext_vector_type(8))) float v8f;

#define NB            16
#define N_ITEMS_C     10000
#define N_SETS_C      1000
#define SET_SZ_C      100
#define DIM_C         128
#define SETS_PER_TILE 16
#define N_TILES       ((N_SETS_C + SETS_PER_TILE - 1) / SETS_PER_TILE)

__global__ __launch_bounds__(256) void mcp_embed_kernel(
    const float* __restrict__ weights,     // [B, N_ITEMS]
    const int*   __restrict__ membership,  // [B, N_SETS, SET_SZ] (int32)
    const float* __restrict__ w,           // [D]
    const float* __restrict__ bias,        // [D]
    float*       __restrict__ out)         // [B, N_SETS, D]
{
    __shared__ float sS[SETS_PER_TILE];

    const int tile = blockIdx.x;           // set tile within batch
    const int b    = blockIdx.y;           // batch
    const int tid  = threadIdx.x;
    const int lane = tid & 31;
    const int wave = tid >> 5;             // 8 waves per block

    // Overlap trick: clamp the last tile so every tile is full. The overlap
    // region is recomputed identically by two blocks -> deterministic output,
    // and ALL bounds checks vanish from the kernel.
    int set_base = tile * SETS_PER_TILE;
    if (set_base > N_SETS_C - SETS_PER_TILE) set_base = N_SETS_C - SETS_PER_TILE;

    const float* wrow = weights + (size_t)b * N_ITEMS_C;

    // ---- Phase 1: S[s] = sum_k weights[b, membership[b,s,k]] ----
    // 8 waves x 2 sets. 100 indices = 25 lanes x int4: one b128 load per lane
    // (fully coalesced, 400B row start is 16B-aligned), 4 gathers per lane.
    #pragma unroll
    for (int si = 0; si < 2; ++si) {
        const int s  = wave * 2 + si;
        const int gs = set_base + s;
        float acc = 0.0f;
        if (lane < (SET_SZ_C / 4)) {
            const int4* mp =
                (const int4*)(membership + ((size_t)b * N_SETS_C + gs) * SET_SZ_C);
            const int4 idx = mp[lane];
            acc = wrow[idx.x] + wrow[idx.y] + wrow[idx.z] + wrow[idx.w];
        }
        // wave32 butterfly reduction
        #pragma unroll
        for (int off = 16; off > 0; off >>= 1)
            acc += __shfl_xor(acc, off, 32);
        if (lane == 0) sS[s] = acc;
    }
    __syncthreads();

    // ---- Phase 2: rank-1 expansion via v_wmma_f32_16x16x4_f32 ----
    // D16x16 = A16x4 * B4x16 + C, only K=0 used:
    //   A[:,0] = S[set tile], B[0,:] = w[d tile], C[m,n] = SET_SZ*bias[d_base+n]
    // 32-bit A layout: lanes 0-15 -> M=lane, VGPR0 = K=0 (lanes 16-31 = K=2 -> 0).
    // 32-bit B layout: VGPR0 lanes 0-15 = row K=0, N=lane (upper half K=2 -> 0).
    // C/D layout: VGPR r -> M = r + 8*(lane>=16), N = lane&15.
    const int d_base = wave * 16;          // 8 waves cover D=128
    const int nl     = lane & 15;

    v2f av;
    av.x = (lane < 16) ? sS[lane] : 0.0f;
    av.y = 0.0f;

    v2f bv;
    bv.x = (lane < 16) ? w[d_base + lane] : 0.0f;
    bv.y = 0.0f;

    const float cval = (float)SET_SZ_C * bias[d_base + nl];
    v8f cv;
    #pragma unroll
    for (int r = 0; r < 8; ++r) cv[r] = cval;

    // (neg_a, A, neg_b, B, c_mod, C, reuse_a, reuse_b)
    v8f dv = __builtin_amdgcn_wmma_f32_16x16x4_f32(
        false, av, false, bv, (short)0, cv, false, false);

    // ---- Store: one base address, 8 unconditional immediate-offset stores ----
    const int mbase = (lane >> 4) * 8;     // rows M = mbase .. mbase+7
    float* op = out + ((size_t)b * N_SETS_C + (size_t)(set_base + mbase)) * DIM_C
                    + d_base + nl;
    #pragma unroll
    for (int r = 0; r < 8; ++r)
        op[(size_t)r * DIM_C] = dv[r];     // offset r*512B, inside 24-bit IOFFSET
}

extern "C" void kernel_launch(void* const* d_in, const int* in_sizes, int n_in,
                              void* d_out, int out_size, void* d_ws, size_t ws_size,
                              hipStream_t stream) {
    (void)in_sizes; (void)n_in; (void)d_ws; (void)ws_size; (void)out_size;
    const float* weights    = (const float*)d_in[0];
    const int*   membership = (const int*)d_in[1];
    const float* w          = (const float*)d_in[2];
    const float* bias       = (const float*)d_in[3];
    float*       out        = (float*)d_out;

    dim3 grid(N_TILES, NB);
    dim3 block(256);
    mcp_embed_kernel<<<grid, block, 0, stream>>>(weights, membership, w, bias, out);
}